// ViewRendering_60756607369864
// MI455X (gfx1250) — compile-verified
//
#include <hip/hip_runtime.h>
#include <hip/hip_bf16.h>
#include <stdint.h>

#define BB 24
#define HH 384
#define WW 640
#define HWPX (HH * WW)        // 245760
#define N3  (3 * HWPX)        // 737280
#define TPB 256
#define PXT 4                 // pixels per thread
#define NBLK (HWPX / (TPB * PXT))  // 240 blocks per batch
#define MT_STRIDE 24          // floats per batch: [0..8]=invK3x3, [9..20]=P(3x4), pad

// ---------------------------------------------------------------------------
// Broadcast 21 per-batch constants into LDS via the CDNA5 async global->LDS
// path (ASYNCcnt). GVS mode: per-lane LDS dest VGPR, per-lane 32-bit offset
// VGPR, SGPR64 base. Low 32 bits of a generic pointer to LDS are the
// wave-relative LDS byte offset (flat aperture mapping).
// ---------------------------------------------------------------------------
__device__ __forceinline__ void load_consts(float* sc, const float* __restrict__ g) {
  if (threadIdx.x < 32) {
    unsigned lane = threadIdx.x;
    if (lane < 21) {
      unsigned ldsoff = (unsigned)(uintptr_t)(sc + lane);
      unsigned goff   = lane * 4u;
      asm volatile("global_load_async_to_lds_b32 %0, %1, %2"
                   :: "v"(ldsoff), "v"(goff), "s"(g)
                   : "memory");
    }
#if __has_builtin(__builtin_amdgcn_s_wait_asynccnt)
    __builtin_amdgcn_s_wait_asynccnt(0);
#else
    asm volatile("s_wait_asynccnt 0" ::: "memory");
#endif
  }
  __syncthreads();
}

__device__ __forceinline__ int clampi(float v, float hi) {
  // NaN-safe: fmaxf(NaN,0)=0
  return (int)fminf(fmaxf(v, 0.0f), hi);
}

// one bilinear tap (3 channels): clip index, gather, multiply by validity
__device__ __forceinline__ void tap3(const float* __restrict__ ib,
                                     float xf, float yf, float out[3]) {
  bool valid = (xf >= 0.0f) && (xf < (float)WW) && (yf >= 0.0f) && (yf < (float)HH);
  int xc = clampi(xf, (float)(WW - 1));
  int yc = clampi(yf, (float)(HH - 1));
  int idx = yc * WW + xc;
  float f = valid ? 1.0f : 0.0f;
  out[0] = ib[idx] * f;
  out[1] = ib[HWPX + idx] * f;
  out[2] = ib[2 * HWPX + idx] * f;
}

// Per-pixel warp: projection + bilinear img sample + nearest mask sample
// + validity. sc: [0..8]=invK3 (row major 3x3), [9..20]=P (row major 3x4).
__device__ __forceinline__ void warp_pixel(const float* __restrict__ img,
                                           const float* __restrict__ msk,
                                           const float* sc, int b, int p, float d,
                                           float iw[3], float& wm) {
  int yy = p / WW;
  int xx = p - yy * WW;
  float xf = (float)xx, yf = (float)yy;

  // cam = invK3 @ (x,y,1);  cam *= depth   (reference op order)
  float c0 = sc[0] * xf + sc[1] * yf + sc[2];
  float c1 = sc[3] * xf + sc[4] * yf + sc[5];
  float c2 = sc[6] * xf + sc[7] * yf + sc[8];
  c0 *= d; c1 *= d; c2 *= d;
  // proj = P @ (cam,1)
  float pr0 = sc[9]  * c0 + sc[10] * c1 + sc[11] * c2 + sc[12];
  float pr1 = sc[13] * c0 + sc[14] * c1 + sc[15] * c2 + sc[16];
  float pr2 = sc[17] * c0 + sc[18] * c1 + sc[19] * c2 + sc[20];

  float den = pr2 + 1e-7f;
  float u = pr0 / den;
  float v = pr1 / den;
  float pcx = u / (float)(WW - 1) * 2.0f - 1.0f;
  float pcy = v / (float)(HH - 1) * 2.0f - 1.0f;

  // grid_sample inverse map (align_corners=True)
  float gx = (pcx + 1.0f) * 0.5f * (float)(WW - 1);
  float gy = (pcy + 1.0f) * 0.5f * (float)(HH - 1);

  // bilinear
  float x0f = floorf(gx), y0f = floorf(gy);
  float wx = gx - x0f, wy = gy - y0f;
  const float* ib = img + (size_t)b * 3 * HWPX;
  float v00[3], v01[3], v10[3], v11[3];
  tap3(ib, x0f,        y0f,        v00);
  tap3(ib, x0f + 1.0f, y0f,        v01);
  tap3(ib, x0f,        y0f + 1.0f, v10);
  tap3(ib, x0f + 1.0f, y0f + 1.0f, v11);
  float w00 = (1.0f - wx) * (1.0f - wy);
  float w01 = wx * (1.0f - wy);
  float w10 = (1.0f - wx) * wy;
  float w11 = wx * wy;
#pragma unroll
  for (int c = 0; c < 3; ++c) {
    float val = v00[c] * w00 + v01[c] * w01 + v10[c] * w10 + v11[c] * w11;
    iw[c] = (val != val) ? 2.0f : val;   // NaN -> 2.0
  }

  // nearest mask sample (round half-to-even like jnp.round)
  float rx = rintf(gx), ry = rintf(gy);
  bool nv = (rx >= 0.0f) && (rx < (float)WW) && (ry >= 0.0f) && (ry < (float)HH);
  int nxc = clampi(rx, (float)(WW - 1));
  int nyc = clampi(ry, (float)(HH - 1));
  float mwv = nv ? msk[(size_t)b * HWPX + nyc * WW + nxc] : 0.0f;
  if (mwv != mwv) mwv = 0.0f;

  bool invalid = (pcx > 1.0f) || (pcx < -1.0f) || (pcy > 1.0f) || (pcy < -1.0f);
  wm = invalid ? 0.0f : mwv;
}

// ---------------------------------------------------------------------------
// Kernel 0: zero accumulators + precompute per-batch invK3x3 and P = (K@T)[:3]
// ---------------------------------------------------------------------------
__global__ __launch_bounds__(256) void init_kernel(const float* __restrict__ invK,
                                                   const float* __restrict__ srcK,
                                                   const float* __restrict__ Tm,
                                                   double* __restrict__ acc,
                                                   float* __restrict__ mt) {
  int t = threadIdx.x;
  if (t < BB * 7) acc[t] = 0.0;
  if (t < BB) {
    const float* K  = srcK + t * 16;
    const float* Tr = Tm   + t * 16;
    const float* iK = invK + t * 16;
    float* o = mt + t * MT_STRIDE;
#pragma unroll
    for (int i = 0; i < 3; ++i)
#pragma unroll
      for (int j = 0; j < 3; ++j) o[i * 3 + j] = iK[i * 4 + j];
#pragma unroll
    for (int i = 0; i < 3; ++i)
#pragma unroll
      for (int j = 0; j < 4; ++j) {
        float s = 0.0f;
#pragma unroll
        for (int k = 0; k < 4; ++k) s += K[i * 4 + k] * Tr[k * 4 + j];
        o[9 + i * 4 + j] = s;
      }
  }
}

// ---------------------------------------------------------------------------
// Kernel 1: fused warp + per-batch statistics, 4 px/thread, b128 streaming.
// acc[b*7 + {0:cnt,1:s_msum,2:s_all,3:s_sq,4:w_msum,5:w_all,6:w_sq}]
// ---------------------------------------------------------------------------
__global__ __launch_bounds__(256) void warp_stats_kernel(const float* __restrict__ img,
                                                         const float* __restrict__ msk,
                                                         const float* __restrict__ dep,
                                                         const float* __restrict__ mt,
                                                         double* __restrict__ acc) {
  __shared__ float sc[32];
  __shared__ float swred[8 * 7];
  int b = blockIdx.y;
  int vi = blockIdx.x * TPB + threadIdx.x;   // float4 index
  int p0 = vi * PXT;

  __builtin_prefetch(dep + (size_t)b * HWPX + p0, 0, 0);  // global_prefetch_b8
  load_consts(sc, mt + b * MT_STRIDE);

  // b128 streaming loads
  size_t b1 = (size_t)b * HWPX;
  size_t b3 = (size_t)b * 3 * HWPX;
  float4 d4  = ((const float4*)(dep + b1))[vi];
  float4 sr4 = ((const float4*)(img + b3))[vi];
  float4 sg4 = ((const float4*)(img + b3 + HWPX))[vi];
  float4 sb4 = ((const float4*)(img + b3 + 2 * HWPX))[vi];
  float4 sm4 = ((const float4*)(msk + b1))[vi];
  float dd[PXT] = {d4.x, d4.y, d4.z, d4.w};
  float rr[PXT] = {sr4.x, sr4.y, sr4.z, sr4.w};
  float gg[PXT] = {sg4.x, sg4.y, sg4.z, sg4.w};
  float bbv[PXT] = {sb4.x, sb4.y, sb4.z, sb4.w};
  float mm[PXT] = {sm4.x, sm4.y, sm4.z, sm4.w};

  float vals[7] = {0, 0, 0, 0, 0, 0, 0};
#pragma unroll
  for (int i = 0; i < PXT; ++i) {
    float iw[3]; float wm;
    warp_pixel(img, msk, sc, b, p0 + i, dd[i], iw, wm);
    float m = (mm[i] * wm > 0.0f) ? 1.0f : 0.0f;
    float ssum = rr[i] + gg[i] + bbv[i];
    float wsum = iw[0] + iw[1] + iw[2];
    vals[0] += m;
    vals[1] += m * ssum;
    vals[2] += ssum;
    vals[3] += rr[i] * rr[i] + gg[i] * gg[i] + bbv[i] * bbv[i];
    vals[4] += m * wsum;
    vals[5] += wsum;
    vals[6] += iw[0] * iw[0] + iw[1] * iw[1] + iw[2] * iw[2];
  }

  // wave32 butterfly reduction (in-register)
#pragma unroll
  for (int off = 16; off > 0; off >>= 1)
#pragma unroll
    for (int k = 0; k < 7; ++k)
      vals[k] += __shfl_xor(vals[k], off, 32);

  int wave = threadIdx.x >> 5;
  int lane = threadIdx.x & 31;
  if (lane == 0)
#pragma unroll
    for (int k = 0; k < 7; ++k) swred[wave * 7 + k] = vals[k];
  __syncthreads();
  if (threadIdx.x < 7) {
    float s = 0.0f;
#pragma unroll
    for (int w = 0; w < 8; ++w) s += swred[w * 7 + threadIdx.x];
    atomicAdd(&acc[b * 7 + threadIdx.x], (double)s);
  }
}

// ---------------------------------------------------------------------------
// Kernel 2: finalize stats per batch + global empty-mask flag
// stats[b*4 + {0:s_mean,1:s_std,2:w_mean,3:w_std}]
// ---------------------------------------------------------------------------
__global__ __launch_bounds__(32) void finalize_kernel(const double* __restrict__ acc,
                                                      float* __restrict__ stats,
                                                      int* __restrict__ flag) {
  int b = threadIdx.x;
  if (b < BB) {
    const double* a = acc + b * 7;
    double cnt = a[0];
    double msum = 3.0 * cnt;
    const double N = (double)N3;
    double smean = a[1] / (msum + 1e-8);
    double svar  = (a[3] - 2.0 * smean * a[2] + N * smean * smean) / N;
    double sstd  = sqrt(svar + 1e-16);
    double wmean = a[4] / (msum + 1e-8);
    double wvar  = (a[6] - 2.0 * wmean * a[5] + N * wmean * wmean) / N;
    double wstd  = sqrt(wvar + 1e-16);
    stats[b * 4 + 0] = (float)smean;
    stats[b * 4 + 1] = (float)sstd;
    stats[b * 4 + 2] = (float)wmean;
    stats[b * 4 + 3] = (float)wstd;
  }
  if (threadIdx.x == 0) {
    int z = 0;
    for (int i = 0; i < BB; ++i)
      if (acc[i * 7] == 0.0) z = 1;
    *flag = z;
  }
}

// ---------------------------------------------------------------------------
// Kernel 3: recompute warp (src_img is L2-resident), normalize, b128 stores.
// ---------------------------------------------------------------------------
__global__ __launch_bounds__(256) void render_kernel(const float* __restrict__ img,
                                                     const float* __restrict__ msk,
                                                     const float* __restrict__ dep,
                                                     const float* __restrict__ mt,
                                                     const float* __restrict__ stats,
                                                     const int* __restrict__ flag,
                                                     float* __restrict__ out) {
  __shared__ float sc[32];
  int b = blockIdx.y;
  int vi = blockIdx.x * TPB + threadIdx.x;
  int p0 = vi * PXT;

  __builtin_prefetch(dep + (size_t)b * HWPX + p0, 0, 0);
  load_consts(sc, mt + b * MT_STRIDE);

  size_t b1 = (size_t)b * HWPX;
  float4 d4 = ((const float4*)(dep + b1))[vi];
  float dd[PXT] = {d4.x, d4.y, d4.z, d4.w};

  float smean = stats[b * 4 + 0];
  float sstd  = stats[b * 4 + 1];
  float wmean = stats[b * 4 + 2];
  float wsp   = stats[b * 4 + 3] + 1e-8f;
  int fz = *flag;

  float och[3][PXT];
#pragma unroll
  for (int i = 0; i < PXT; ++i) {
    float iw[3]; float wm;
    warp_pixel(img, msk, sc, b, p0 + i, dd[i], iw, wm);
#pragma unroll
    for (int c = 0; c < 3; ++c) {
      float nw = (iw[c] - wmean) / wsp * sstd + smean;
      och[c][i] = fz ? iw[c] : nw * wm;
    }
  }

  size_t ob = (size_t)b * 3 * HWPX;
#pragma unroll
  for (int c = 0; c < 3; ++c) {
    float4 o4 = make_float4(och[c][0], och[c][1], och[c][2], och[c][3]);
    ((float4*)(out + ob + (size_t)c * HWPX))[vi] = o4;
  }
}

// ---------------------------------------------------------------------------
extern "C" void kernel_launch(void* const* d_in, const int* in_sizes, int n_in,
                              void* d_out, int out_size, void* d_ws, size_t ws_size,
                              hipStream_t stream) {
  (void)in_sizes; (void)n_in; (void)out_size; (void)ws_size;
  const float* src_img  = (const float*)d_in[0];
  const float* src_mask = (const float*)d_in[1];
  const float* tar_dep  = (const float*)d_in[2];
  const float* tar_invK = (const float*)d_in[3];
  const float* src_K    = (const float*)d_in[4];
  const float* Tmat     = (const float*)d_in[5];
  float* out = (float*)d_out;

  // workspace layout: doubles first (alignment), then floats, then flag
  double* acc  = (double*)d_ws;                                   // BB*7
  float*  mt   = (float*)((char*)d_ws + BB * 7 * sizeof(double)); // BB*MT_STRIDE
  float*  stat = mt + BB * MT_STRIDE;                             // BB*4
  int*    flag = (int*)(stat + BB * 4);

  dim3 grid(NBLK, BB);
  init_kernel<<<1, 256, 0, stream>>>(tar_invK, src_K, Tmat, acc, mt);
  warp_stats_kernel<<<grid, TPB, 0, stream>>>(src_img, src_mask, tar_dep, mt, acc);
  finalize_kernel<<<1, 32, 0, stream>>>(acc, stat, flag);
  render_kernel<<<grid, TPB, 0, stream>>>(src_img, src_mask, tar_dep, mt, stat, flag, out);
}